// GraphAttention_79190607004093
// MI455X (gfx1250) — compile-verified
//
#include <hip/hip_runtime.h>
#include <math.h>

#define NNODES 100000
#define NEDGES 1600000
#define DIN 128
#define DOUT 64
#define NHEADS 4
#define LRELU_ALPHA 0.2f

typedef __attribute__((ext_vector_type(2))) float v2f;
typedef __attribute__((ext_vector_type(4))) float v4f;
typedef __attribute__((ext_vector_type(8))) float v8f;

__device__ __forceinline__ float eluf(float x) {
    return x > 0.f ? x : (__expf(x) - 1.f);
}

// ---------------------------------------------------------------------------
// Kernel 1: CSR row pointers from sorted edge_row via per-row binary search.
// ---------------------------------------------------------------------------
__global__ __launch_bounds__(256) void gat_row_ptr(const int* __restrict__ edge_row,
                                                   int* __restrict__ row_ptr) {
    int r = blockIdx.x * blockDim.x + threadIdx.x;
    if (r > NNODES) return;
    int lo = 0, hi = NEDGES;
    while (lo < hi) {
        int mid = (lo + hi) >> 1;
        if (edge_row[mid] < r) lo = mid + 1; else hi = mid;
    }
    row_ptr[r] = lo;
}

// ---------------------------------------------------------------------------
// Kernel 2: per-head projection xh[h] = inputs @ W[h] using fp32 WMMA.
// Block = 128 threads (4 waves). Each wave owns a 16-row tile and computes
// all 64 output columns (4 accumulators), reusing one A fragment per K-step.
// W[h] is staged in LDS packed as K-pairs: sW2[k/2][n] = (W[k][n], W[k+1][n]),
// so every B fragment is a single aligned ds_load_b64 straight into the
// even-aligned VGPR pair WMMA requires (no register shuffling).
// ---------------------------------------------------------------------------
__global__ __launch_bounds__(128) void gat_proj_wmma(const float* __restrict__ inputs,
                                                     const float* __restrict__ W,
                                                     float* __restrict__ xh) {
    __shared__ v2f sW2[(DIN / 2) * DOUT];  // 32 KB, K-pair packed
    const int h   = blockIdx.y;
    const int tid = threadIdx.x;
    const float* Wh = W + (size_t)h * DIN * DOUT;

    // Cooperative K-pair packing stage: i -> (k2 = i/64, n = i%64).
    // Consecutive threads touch consecutive n -> coalesced global loads.
    for (int i = tid; i < (DIN / 2) * DOUT; i += 128) {
        const int k2 = i >> 6;
        const int n  = i & 63;
        v2f p;
        p.x = Wh[(2 * k2) * DOUT + n];
        p.y = Wh[(2 * k2 + 1) * DOUT + n];
        sW2[i] = p;
    }
    __syncthreads();

    const int wave = tid >> 5;
    const int lane = tid & 31;
    const int rowTile = blockIdx.x * 4 + wave;          // 16-row tile index
    const int m    = lane & 15;                         // row within tile
    const int kgrp = (lane >> 4) * 2;                   // K group: 0 or 2

    long long row  = (long long)rowTile * 16 + m;
    long long rowc = row < NNODES ? row : (NNODES - 1); // clamp OOB loads
    const float* Arow = inputs + rowc * DIN;

    v8f c0 = {}, c1 = {}, c2 = {}, c3 = {};
    const int ncol = lane & 15;
    // Lane's base into the packed LDS: k-pair index = kgrp/2, column = ncol.
    const v2f* Bbase = sW2 + (kgrp >> 1) * DOUT + ncol;

    for (int kk = 0; kk < DIN; kk += 4) {
        const int k0 = kk + kgrp;
        // A fragment (16x4 f32): lane holds A[m][k0], A[m][k0+1] (contiguous).
        v2f a = *(const v2f*)(Arow + k0);
        // B fragments (4x16 f32): one ds_load_b64 per column tile.
        const v2f* Bk = Bbase + (kk >> 1) * DOUT;
        v2f b0 = Bk[0];
        v2f b1 = Bk[16];
        v2f b2 = Bk[32];
        v2f b3 = Bk[48];
        c0 = __builtin_amdgcn_wmma_f32_16x16x4_f32(false, a, false, b0, (short)0, c0, false, false);
        c1 = __builtin_amdgcn_wmma_f32_16x16x4_f32(false, a, false, b1, (short)0, c1, false, false);
        c2 = __builtin_amdgcn_wmma_f32_16x16x4_f32(false, a, false, b2, (short)0, c2, false, false);
        c3 = __builtin_amdgcn_wmma_f32_16x16x4_f32(false, a, false, b3, (short)0, c3, false, false);
    }

    // C/D layout (f32 16x16): VGPR v -> lanes 0-15: M=v, lanes 16-31: M=v+8; N=lane&15.
    const int mofs = (lane >> 4) * 8;
    float* outh = xh + (size_t)h * NNODES * DOUT;
    for (int v = 0; v < 8; ++v) {
        long long mrow = (long long)rowTile * 16 + mofs + v;
        if (mrow < NNODES) {
            float* orow = outh + mrow * DOUT + ncol;
            orow[0]  = c0[v];
            orow[16] = c1[v];
            orow[32] = c2[v];
            orow[48] = c3[v];
        }
    }
}

// ---------------------------------------------------------------------------
// Kernel 3: f1 = elu(xh @ a1), f2 = elu(xh @ a2), per (head, node).
// ---------------------------------------------------------------------------
__global__ __launch_bounds__(256) void gat_f12(const float* __restrict__ xh,
                                               const float* __restrict__ a1,
                                               const float* __restrict__ a2,
                                               float* __restrict__ f1,
                                               float* __restrict__ f2) {
    int idx = blockIdx.x * blockDim.x + threadIdx.x;  // idx = h*N + n
    if (idx >= NHEADS * NNODES) return;
    int h = idx / NNODES;
    const v4f* xv  = (const v4f*)(xh + (size_t)idx * DOUT);
    const v4f* a1v = (const v4f*)(a1 + h * DOUT);
    const v4f* a2v = (const v4f*)(a2 + h * DOUT);
    float s1 = 0.f, s2 = 0.f;
    for (int i = 0; i < DOUT / 4; ++i) {
        v4f x = xv[i], u = a1v[i], w = a2v[i];
        s1 += x.x * u.x + x.y * u.y + x.z * u.z + x.w * u.w;
        s2 += x.x * w.x + x.y * w.y + x.z * w.z + x.w * w.w;
    }
    f1[idx] = eluf(s1);
    f2[idx] = eluf(s2);
}

// ---------------------------------------------------------------------------
// Kernel 4: one wave32 per destination row. Wave-parallel segment max / sum
// of exp (softmax), then weighted aggregation of xh[col] rows with the 32
// lanes covering the 64 output dims (2 each). Heads fused; elu + mean at end.
// ---------------------------------------------------------------------------
__global__ __launch_bounds__(256) void gat_attn(const float* __restrict__ xh,
                                                const float* __restrict__ f1,
                                                const float* __restrict__ f2,
                                                const float* __restrict__ b,
                                                const float* __restrict__ edge_val,
                                                const int* __restrict__ edge_col,
                                                const int* __restrict__ row_ptr,
                                                float* __restrict__ out) {
    const int wid  = (blockIdx.x * blockDim.x + threadIdx.x) >> 5;
    const int lane = threadIdx.x & 31;
    if (wid >= NNODES) return;
    const int r  = wid;
    const int e0 = row_ptr[r];
    const int e1 = row_ptr[r + 1];

    float res0 = 0.f, res1 = 0.f;
    for (int h = 0; h < NHEADS; ++h) {
        const float* f1h = f1 + (size_t)h * NNODES;
        const float* f2h = f2 + (size_t)h * NNODES;
        const float* xhh = xh + (size_t)h * NNODES * DOUT;
        float acc0 = 0.f, acc1 = 0.f;
        if (e1 > e0) {
            const float f1r = f1h[r];
            // Pass 1: segment max of leaky_relu scores.
            float m = -3.0e38f;
            for (int e = e0 + lane; e < e1; e += 32) {
                float le = edge_val[e] * (f1r + f2h[edge_col[e]]);
                le = le > 0.f ? le : LRELU_ALPHA * le;
                m = fmaxf(m, le);
            }
            for (int off = 16; off > 0; off >>= 1)
                m = fmaxf(m, __shfl_xor(m, off, 32));
            // Pass 2: exp-sum + weighted gather-accumulate in chunks of 32.
            float s = 0.f;
            for (int base = e0; base < e1; base += 32) {
                const int e = base + lane;
                float ex = 0.f; int c = 0;
                if (e < e1) {
                    c = edge_col[e];
                    float le = edge_val[e] * (f1r + f2h[c]);
                    le = le > 0.f ? le : LRELU_ALPHA * le;
                    ex = __expf(le - m);
                }
                s += ex;
                const int nch = min(32, e1 - base);
                for (int j = 0; j < nch; ++j) {
                    const float exj = __shfl(ex, j, 32);
                    const int   cj  = __shfl(c, j, 32);
                    const float* xrow = xhh + (size_t)cj * DOUT;
                    acc0 += exj * xrow[lane];
                    acc1 += exj * xrow[lane + 32];
                }
            }
            for (int off = 16; off > 0; off >>= 1)
                s += __shfl_xor(s, off, 32);
            const float inv = 1.f / s;
            acc0 *= inv; acc1 *= inv;
        }
        const float* bh = b + h * DOUT;
        res0 += eluf(acc0 + bh[lane]);
        res1 += eluf(acc1 + bh[lane + 32]);
    }
    float* orow = out + (size_t)r * DOUT;
    orow[lane]      = res0 * (1.f / NHEADS);
    orow[lane + 32] = res1 * (1.f / NHEADS);
}

// ---------------------------------------------------------------------------
extern "C" void kernel_launch(void* const* d_in, const int* in_sizes, int n_in,
                              void* d_out, int out_size, void* d_ws, size_t ws_size,
                              hipStream_t stream) {
    const float* inputs   = (const float*)d_in[0];
    const float* W        = (const float*)d_in[1];
    const float* a1       = (const float*)d_in[2];
    const float* a2       = (const float*)d_in[3];
    const float* b        = (const float*)d_in[4];
    const float* edge_val = (const float*)d_in[5];
    const int*   edge_row = (const int*)d_in[6];
    const int*   edge_col = (const int*)d_in[7];
    float*       out      = (float*)d_out;

    char* ws = (char*)d_ws;
    float* xh = (float*)ws;
    size_t off = (size_t)NHEADS * NNODES * DOUT * sizeof(float);       // 102.4 MB
    float* f1 = (float*)(ws + off); off += (size_t)NHEADS * NNODES * sizeof(float);
    float* f2 = (float*)(ws + off); off += (size_t)NHEADS * NNODES * sizeof(float);
    int* row_ptr = (int*)(ws + off);

    // 1) CSR row pointers (independent of projection).
    gat_row_ptr<<<(NNODES + 1 + 255) / 256, 256, 0, stream>>>(edge_row, row_ptr);

    // 2) WMMA projection: grid.x covers 64-row super-tiles, grid.y = heads.
    dim3 gProj((NNODES + 63) / 64, NHEADS);
    gat_proj_wmma<<<gProj, 128, 0, stream>>>(inputs, W, xh);

    // 3) Per-node attention logit halves.
    gat_f12<<<(NHEADS * NNODES + 255) / 256, 256, 0, stream>>>(xh, a1, a2, f1, f2);

    // 4) Softmax + aggregation: one wave per row, 8 waves per block.
    gat_attn<<<(NNODES + 7) / 8, 256, 0, stream>>>(xh, f1, f2, b, edge_val,
                                                   edge_col, row_ptr, out);
}